// gnn_38714835206709
// MI455X (gfx1250) — compile-verified
//
#include <hip/hip_runtime.h>
#include <math.h>

// Problem constants (match reference)
#define B_    128
#define N1_   128
#define N2_   384
#define N_    512
#define E_    1024
#define DIN_  64
#define DFC_  128

typedef __attribute__((ext_vector_type(2))) float v2f;
typedef __attribute__((ext_vector_type(4))) float v4f;
typedef __attribute__((ext_vector_type(8))) float v8f;

// ---------------------------------------------------------------- reductions
__device__ __forceinline__ float blockReduceSum(float v, float* red) {
  const int t = threadIdx.x;
  red[t] = v;
  __syncthreads();
  for (int s = blockDim.x >> 1; s > 0; s >>= 1) {
    if (t < s) red[t] += red[t + s];
    __syncthreads();
  }
  float r = red[0];
  __syncthreads();
  return r;
}

// ---------------------------------------------------------------- init
__global__ void init_kernel(float* __restrict__ out, float* __restrict__ topb,
                            float* __restrict__ botb) {
  const int t = threadIdx.x;  // 256
  if (t < B_) topb[t] = 0.f;
  else        botb[t - B_] = 0.f;
  if (t < 4)  out[B_ + t] = 0.f;
}

// ---------------------------------------------------------------- pooled = (c_hs * c_valid).sum(1)/N1
__global__ void pooled_kernel(const float* __restrict__ c_hs, const float* __restrict__ c_valid,
                              float* __restrict__ pooled) {
  const int b = blockIdx.x, d = threadIdx.x;  // 64 threads, coalesced on d
  const float* base = c_hs + (size_t)b * N_ * DIN_ + d;
  const float* cv   = c_valid + (size_t)b * N_;
  float s = 0.f;
  for (int n = 0; n < N_; ++n) s = fmaf(base[(size_t)n * DIN_], cv[n], s);
  pooled[b * DIN_ + d] = s * (1.0f / (float)N1_);
}

// ---------------------------------------------------------------- WMMA f32 GEMM: C(128x128) = relu(A(128xK) @ B(Kx128) + bias)
// One wave (32 lanes) per 16x16 tile; V_WMMA_F32_16X16X4_F32, K-loop step 4.
// A frag (16x4):  m = lane&15, k = k0 + 2*(lane>>4) .. +1
// B frag (4x16):  n = lane&15, k = k0 + 2*(lane>>4) .. +1
// C/D (16x16):    vgpr r -> M = r + 8*(lane>>4), N = lane&15
__global__ void mlp_gemm_wmma(const float* __restrict__ A, const float* __restrict__ Bm,
                              const float* __restrict__ bias, float* __restrict__ C, int K) {
  const int lane  = threadIdx.x;      // 32 = 1 wave
  const int tileN = blockIdx.x;       // 0..7
  const int tileM = blockIdx.y;       // 0..7
  const int half  = lane >> 4;
  const int l15   = lane & 15;
  const int rowA  = tileM * 16 + l15;
  const int colB  = tileN * 16 + l15;
  v8f acc = {};
  for (int k0 = 0; k0 < K; k0 += 4) {
    const int ka = k0 + half * 2;
    v2f a, b;
    a.x = A[rowA * K + ka];
    a.y = A[rowA * K + ka + 1];
    b.x = Bm[ka * DFC_ + colB];
    b.y = Bm[(ka + 1) * DFC_ + colB];
    acc = __builtin_amdgcn_wmma_f32_16x16x4_f32(false, a, false, b, (short)0, acc,
                                                false, false);
  }
  const float bb = bias[colB];
#pragma unroll
  for (int r = 0; r < 8; ++r) {
    const int row = tileM * 16 + half * 8 + r;
    C[row * DFC_ + colB] = fmaxf(acc[r] + bb, 0.f);
  }
}

// ---------------------------------------------------------------- scores = sigmoid(h2 @ W3 + b3)
__global__ void score_kernel(const float* __restrict__ h2, const float* __restrict__ W3,
                             const float* __restrict__ b3, float* __restrict__ out) {
  const int b = threadIdx.x;  // 128
  float s = b3[0];
  for (int d = 0; d < DFC_; ++d) s = fmaf(h2[b * DFC_ + d], W3[d], s);
  out[b] = 1.f / (1.f + __expf(-s));
}

// ---------------------------------------------------------------- column norms of atten
// nrm[b*512 + j]: j<128 -> ||attention[b,j,:]|| (row norm), j>=128 -> ||attention[b,:,j-128]|| (col norm)
__global__ void norms_kernel(const float* __restrict__ att, float* __restrict__ nrm) {
  const int b = blockIdx.x, t = threadIdx.x;  // 512 threads
  float s = 0.f;
  if (t < N1_) {
    const float* row = att + ((size_t)b * N1_ + t) * N2_;
    for (int k = 0; k < N2_; ++k) { const float v = row[k]; s = fmaf(v, v, s); }
  } else {
    const float* base = att + (size_t)b * N1_ * N2_ + (t - N1_);
    for (int i = 0; i < N1_; ++i) { const float v = base[(size_t)i * N2_]; s = fmaf(v, v, s); }
  }
  nrm[(size_t)b * N_ + t] = fmaxf(sqrtf(s), 1e-12f);
}

// ---------------------------------------------------------------- fused attn_loss pass (BW-critical)
// One block per atten row (b,i). mapping/samelb are 268 MB combined (> 192 MB L2) and
// read exactly once -> stream them with NON-TEMPORAL b128 loads so the thrice-reused
// 25 MB `attention` tensor stays L2-resident. atten value reconstructed analytically.
__global__ void attn_main_kernel(const float* __restrict__ att, const float* __restrict__ mapping,
                                 const float* __restrict__ samelb, const float* __restrict__ nrm,
                                 float* __restrict__ topb, float* __restrict__ botb) {
  const int i = blockIdx.x;   // 0..511
  const int b = blockIdx.y;   // 0..127
  const int t = threadIdx.x;  // 128 -> j = 4t..4t+3
  const size_t rowoff = ((size_t)b * N_ + i) * N_;
  const v4f m4 = __builtin_nontemporal_load(((const v4f*)(mapping + rowoff)) + t);
  const v4f s4 = __builtin_nontemporal_load(((const v4f*)(samelb + rowoff)) + t);
  float av[4] = {0.f, 0.f, 0.f, 0.f};
  const int j0 = t * 4;
  if (i < N1_) {
    if (j0 >= N1_) {
      const v4f at4 = ((const v4f*)(att + ((size_t)b * N1_ + i) * N2_))[t - N1_ / 4];
      const v4f n4  = ((const v4f*)(nrm + (size_t)b * N_))[t];
      av[0] = at4.x / n4.x; av[1] = at4.y / n4.y; av[2] = at4.z / n4.z; av[3] = at4.w / n4.w;
    }
  } else {
    if (j0 < N1_) {
      const float* ab = att + (size_t)b * N1_ * N2_ + (i - N1_);  // attention[b, j, i-128]
      const v4f n4 = ((const v4f*)(nrm + (size_t)b * N_))[t];
      av[0] = ab[(size_t)(j0 + 0) * N2_] / n4.x;
      av[1] = ab[(size_t)(j0 + 1) * N2_] / n4.y;
      av[2] = ab[(size_t)(j0 + 2) * N2_] / n4.z;
      av[3] = ab[(size_t)(j0 + 3) * N2_] / n4.w;
    }
  }
  const float mv[4] = {m4.x, m4.y, m4.z, m4.w};
  const float sv[4] = {s4.x, s4.y, s4.z, s4.w};
  float top = 0.f, bot = 0.f;
#pragma unroll
  for (int q = 0; q < 4; ++q) {
    if (mv[q] > 0.5f || sv[q] > 0.5f) {  // masks are exact 0/1
      const float e = __expf(-av[q]);
      if (mv[q] > 0.5f) top += e;
      if (sv[q] > 0.5f) bot += e;
    }
  }
  __shared__ float red[128];
  const float tsum = blockReduceSum(top, red);
  const float bsum = blockReduceSum(bot, red);
  if (t == 0) {
    atomicAdd(&topb[b], tsum);
    atomicAdd(&botb[b], bsum);
  }
}

__global__ void attn_final_kernel(const float* __restrict__ topb, const float* __restrict__ botb,
                                  float* __restrict__ out) {
  __shared__ float red[128];
  const int t = threadIdx.x;  // 128
  const float v = topb[t] / (botb[t] - topb[t] + 1.0f);
  const float s = blockReduceSum(v, red);
  if (t == 0) out[B_] = s * (1.0f / (float)B_);  // THETA = 1
}

// ---------------------------------------------------------------- Kabsch per batch (block = 128 threads)
__global__ void kabsch_kernel(const float* __restrict__ att, const float* __restrict__ coords,
                              const float* __restrict__ upd, const float* __restrict__ nm,
                              float* __restrict__ out) {
  const int b = blockIdx.x;
  const int t = threadIdx.x;  // 128 = one point per thread
  __shared__ float red[128];
  __shared__ float Rm[9];
  __shared__ float Tv[3];

  // Qsel: forward value of straight-through one-hot == row argmax gather
  const float* arow = att + ((size_t)b * N1_ + t) * N2_;
  float best = arow[0]; int bi = 0;
  for (int k = 1; k < N2_; ++k) { const float v = arow[k]; if (v > best) { best = v; bi = k; } }

  const float* pb = upd + ((size_t)b * N_ + t) * 3;
  const float px = pb[0], py = pb[1], pz = pb[2];
  const float* qb = coords + ((size_t)b * N_ + N1_ + bi) * 3;
  const float qx = qb[0], qy = qb[1], qz = qb[2];
  const float mk = (nm[b * N1_ + t] > 0.5f) ? 1.f : 0.f;

  const float inv = 1.0f / (float)N1_;
  const float Pmx = blockReduceSum(px, red) * inv;
  const float Pmy = blockReduceSum(py, red) * inv;
  const float Pmz = blockReduceSum(pz, red) * inv;
  const float Qmx = blockReduceSum(qx, red) * inv;
  const float Qmy = blockReduceSum(qy, red) * inv;
  const float Qmz = blockReduceSum(qz, red) * inv;
  const float mcnt = blockReduceSum(mk, red);

  const float dpx = px - Pmx, dpy = py - Pmy, dpz = pz - Pmz;
  const float dqx = qx - Qmx, dqy = qy - Qmy, dqz = qz - Qmz;
  float h[9];
  h[0] = blockReduceSum(dpx * dqx, red);
  h[1] = blockReduceSum(dpx * dqy, red);
  h[2] = blockReduceSum(dpx * dqz, red);
  h[3] = blockReduceSum(dpy * dqx, red);
  h[4] = blockReduceSum(dpy * dqy, red);
  h[5] = blockReduceSum(dpy * dqz, red);
  h[6] = blockReduceSum(dpz * dqx, red);
  h[7] = blockReduceSum(dpz * dqy, red);
  h[8] = blockReduceSum(dpz * dqz, red);

  if (t == 0) {
    float A[9];
    for (int k = 0; k < 9; ++k) A[k] = h[k] * (1.0f / 8.0f);  // / NUM_HEADS
    // M = A^T A (symmetric), Jacobi eigendecomposition -> V, eigenvalues
    float M[9];
    for (int r = 0; r < 3; ++r)
      for (int c = 0; c < 3; ++c)
        M[r * 3 + c] = A[0 * 3 + r] * A[0 * 3 + c] + A[1 * 3 + r] * A[1 * 3 + c] +
                       A[2 * 3 + r] * A[2 * 3 + c];
    float V[9] = {1, 0, 0, 0, 1, 0, 0, 0, 1};
    for (int sweep = 0; sweep < 12; ++sweep) {
      for (int pq = 0; pq < 3; ++pq) {
        const int p = (pq == 2) ? 1 : 0;
        const int q = (pq == 0) ? 1 : 2;
        const float apq = M[p * 3 + q];
        if (fabsf(apq) < 1e-30f) continue;
        const float app = M[p * 3 + p], aqq = M[q * 3 + q];
        const float theta = 0.5f * (aqq - app) / apq;
        const float tt = copysignf(1.f, theta) / (fabsf(theta) + sqrtf(theta * theta + 1.f));
        const float cc = rsqrtf(tt * tt + 1.f);
        const float ss = tt * cc;
        M[p * 3 + p] = app - tt * apq;
        M[q * 3 + q] = aqq + tt * apq;
        M[p * 3 + q] = 0.f; M[q * 3 + p] = 0.f;
        const int k = 3 - p - q;
        const float mkp = M[k * 3 + p], mkq = M[k * 3 + q];
        M[k * 3 + p] = cc * mkp - ss * mkq;  M[p * 3 + k] = M[k * 3 + p];
        M[k * 3 + q] = ss * mkp + cc * mkq;  M[q * 3 + k] = M[k * 3 + q];
        for (int r = 0; r < 3; ++r) {
          const float vp = V[r * 3 + p], vq = V[r * 3 + q];
          V[r * 3 + p] = cc * vp - ss * vq;
          V[r * 3 + q] = ss * vp + cc * vq;
        }
      }
    }
    float ev[3] = {M[0], M[4], M[8]};
    int idx[3] = {0, 1, 2};
    if (ev[idx[0]] < ev[idx[1]]) { const int x = idx[0]; idx[0] = idx[1]; idx[1] = x; }
    if (ev[idx[0]] < ev[idx[2]]) { const int x = idx[0]; idx[0] = idx[2]; idx[2] = x; }
    if (ev[idx[1]] < ev[idx[2]]) { const int x = idx[1]; idx[1] = idx[2]; idx[2] = x; }
    float Vs[9], U[9], svv[3];
    for (int i = 0; i < 3; ++i) {
      const int c = idx[i];
      svv[i] = sqrtf(fmaxf(ev[c], 0.f));
      for (int r = 0; r < 3; ++r) Vs[r * 3 + i] = V[r * 3 + c];
    }
    for (int i = 0; i < 3; ++i) {  // U cols = A v_i / s_i
      float ux = A[0] * Vs[0 + i] + A[1] * Vs[3 + i] + A[2] * Vs[6 + i];
      float uy = A[3] * Vs[0 + i] + A[4] * Vs[3 + i] + A[5] * Vs[6 + i];
      float uz = A[6] * Vs[0 + i] + A[7] * Vs[3 + i] + A[8] * Vs[6 + i];
      const float si = svv[i];
      if (si > 1e-20f) { ux /= si; uy /= si; uz /= si; }
      U[0 + i] = ux; U[3 + i] = uy; U[6 + i] = uz;
    }
    if (svv[2] <= 1e-20f) {  // degenerate: u2 = u0 x u1
      U[0 + 2] = U[3 + 0] * U[6 + 1] - U[6 + 0] * U[3 + 1];
      U[3 + 2] = U[6 + 0] * U[0 + 1] - U[0 + 0] * U[6 + 1];
      U[6 + 2] = U[0 + 0] * U[3 + 1] - U[3 + 0] * U[0 + 1];
    }
    const float det = A[0] * (A[4] * A[8] - A[5] * A[7]) -
                      A[1] * (A[3] * A[8] - A[5] * A[6]) +
                      A[2] * (A[3] * A[7] - A[4] * A[6]);
    const float d = (det >= 0.f) ? 1.f : -1.f;
    for (int r = 0; r < 3; ++r)
      for (int c = 0; c < 3; ++c)
        Rm[r * 3 + c] = U[r * 3 + 0] * Vs[c * 3 + 0] + U[r * 3 + 1] * Vs[c * 3 + 1] +
                        d * U[r * 3 + 2] * Vs[c * 3 + 2];
    Tv[0] = Qmx - (Rm[0] * Pmx + Rm[1] * Pmy + Rm[2] * Pmz);
    Tv[1] = Qmy - (Rm[3] * Pmx + Rm[4] * Pmy + Rm[5] * Pmz);
    Tv[2] = Qmz - (Rm[6] * Pmx + Rm[7] * Pmy + Rm[8] * Pmz);
  }
  __syncthreads();
  const float ppx = Rm[0] * px + Rm[1] * py + Rm[2] * pz + Tv[0];
  const float ppy = Rm[3] * px + Rm[4] * py + Rm[5] * pz + Tv[1];
  const float ppz = Rm[6] * px + Rm[7] * py + Rm[8] * pz + Tv[2];
  const float ex = ppx - qx, ey = ppy - qy, ez = ppz - qz;
  const float errsum = blockReduceSum(mk * (ex * ex + ey * ey + ez * ez), red);
  const float mpx = blockReduceSum(ppx, red) * inv;
  const float mpy = blockReduceSum(ppy, red) * inv;
  const float mpz = blockReduceSum(ppz, red) * inv;
  if (t == 0) {
    const float cnt = fmaxf(mcnt * 3.f, 1.f);
    const float rmsd = errsum / cnt;
    const float cx = mpx - Qmx, cy = mpy - Qmy, cz = mpz - Qmz;
    const float cen = (cx * cx + cy * cy + cz * cz) * (1.f / 3.f);
    atomicAdd(out + B_ + 1, rmsd * (1.f / (float)B_));
    atomicAdd(out + B_ + 3, cen * (1.f / (float)B_));
  }
}

// ---------------------------------------------------------------- pairdist loss
__global__ void pairdst_kernel(const int* __restrict__ edge, const float* __restrict__ coords,
                               const float* __restrict__ upd, float* __restrict__ out) {
  const int b = blockIdx.x;
  const int t = threadIdx.x;  // 256
  float acc = 0.f;
  for (int e = t; e < E_; e += 256) {
    const int i0 = edge[((size_t)b * E_ + e) * 2 + 0];
    const int i1 = edge[((size_t)b * E_ + e) * 2 + 1];
    const float* c0 = coords + ((size_t)b * N_ + i0) * 3;
    const float* c1 = coords + ((size_t)b * N_ + i1) * 3;
    const float* u0 = upd + ((size_t)b * N_ + i0) * 3;
    const float* u1 = upd + ((size_t)b * N_ + i1) * 3;
    float dx = c0[0] - c1[0], dy = c0[1] - c1[1], dz = c0[2] - c1[2];
    const float d0 = sqrtf(dx * dx + dy * dy + dz * dz + 1e-12f);
    dx = u0[0] - u1[0]; dy = u0[1] - u1[1]; dz = u0[2] - u1[2];
    const float d1 = sqrtf(dx * dx + dy * dy + dz * dz + 1e-12f);
    acc += d1 - d0;
  }
  __shared__ float red[256];
  const float s = blockReduceSum(acc, red);
  if (t == 0) atomicAdd(out + B_ + 2, fabsf(s) * (1.f / (float)B_));
}

// ---------------------------------------------------------------- launch
extern "C" void kernel_launch(void* const* d_in, const int* in_sizes, int n_in,
                              void* d_out, int out_size, void* d_ws, size_t ws_size,
                              hipStream_t stream) {
  const float* c_hs      = (const float*)d_in[0];
  const float* attention = (const float*)d_in[1];
  const float* coords    = (const float*)d_in[2];
  const float* upd       = (const float*)d_in[3];
  const float* c_valid   = (const float*)d_in[4];
  const float* nm        = (const float*)d_in[5];
  const float* mapping   = (const float*)d_in[6];
  const float* samelb    = (const float*)d_in[7];
  const int*   edge      = (const int*)d_in[8];
  const float* W0 = (const float*)d_in[9];
  const float* b0 = (const float*)d_in[10];
  const float* W1 = (const float*)d_in[11];
  const float* b1 = (const float*)d_in[12];
  const float* W2 = (const float*)d_in[13];
  const float* b2 = (const float*)d_in[14];
  const float* W3 = (const float*)d_in[15];
  const float* b3 = (const float*)d_in[16];
  float* out = (float*)d_out;

  float* ws     = (float*)d_ws;
  float* nrm    = ws;                           // B*N          = 65536
  float* topb   = nrm + (size_t)B_ * N_;        // B
  float* botb   = topb + B_;                    // B
  float* pooled = botb + B_;                    // B*DIN        = 8192
  float* h0     = pooled + (size_t)B_ * DIN_;   // B*DFC        = 16384
  float* h1     = h0 + (size_t)B_ * DFC_;
  float* h2     = h1 + (size_t)B_ * DFC_;

  hipLaunchKernelGGL(init_kernel,      dim3(1),        dim3(256), 0, stream, out, topb, botb);
  hipLaunchKernelGGL(pooled_kernel,    dim3(B_),       dim3(DIN_), 0, stream, c_hs, c_valid, pooled);
  hipLaunchKernelGGL(mlp_gemm_wmma,    dim3(8, 8),     dim3(32),  0, stream, pooled, W0, b0, h0, DIN_);
  hipLaunchKernelGGL(mlp_gemm_wmma,    dim3(8, 8),     dim3(32),  0, stream, h0, W1, b1, h1, DFC_);
  hipLaunchKernelGGL(mlp_gemm_wmma,    dim3(8, 8),     dim3(32),  0, stream, h1, W2, b2, h2, DFC_);
  hipLaunchKernelGGL(score_kernel,     dim3(1),        dim3(B_),  0, stream, h2, W3, b3, out);
  hipLaunchKernelGGL(norms_kernel,     dim3(B_),       dim3(N_),  0, stream, attention, nrm);
  hipLaunchKernelGGL(attn_main_kernel, dim3(N_, B_),   dim3(128), 0, stream, attention, mapping, samelb, nrm, topb, botb);
  hipLaunchKernelGGL(attn_final_kernel,dim3(1),        dim3(B_),  0, stream, topb, botb, out);
  hipLaunchKernelGGL(kabsch_kernel,    dim3(B_),       dim3(N1_), 0, stream, attention, coords, upd, nm, out);
  hipLaunchKernelGGL(pairdst_kernel,   dim3(B_),       dim3(256), 0, stream, edge, coords, upd, out);

  (void)in_sizes; (void)n_in; (void)out_size; (void)ws_size;
}